// QuantumGatesReal_22643067585264
// MI455X (gfx1250) — compile-verified
//
#include <hip/hip_runtime.h>
#include <math.h>

#define NQ      12
#define DEPTH   4
#define DIM     4096
#define NBATCH  4096
#define THREADS 256
#define PERTH   (DIM / THREADS)   // 16 amplitudes per thread

typedef float v2f __attribute__((ext_vector_type(2)));
typedef float v8f __attribute__((ext_vector_type(8)));

// ---------------------------------------------------------------------------
// CNOT ladder (0,1)(1,2)...(10,11)(11,0) as a GF(2)-linear index map.
// qubit q <-> bit (11-q). state'[permL(i)] = state[i].
// ---------------------------------------------------------------------------
__device__ __forceinline__ unsigned permL(unsigned i) {
#pragma unroll
    for (int k = 10; k >= 0; --k)          // CNOT(q,q+1): ctrl bit k+1 -> tgt bit k
        i ^= ((i >> (k + 1)) & 1u) << k;
    i ^= (i & 1u) << 11;                   // CNOT(11,0): ctrl bit 0 -> tgt bit 11
    return i;
}

// ---------------------------------------------------------------------------
// Setup kernel: fuse Rz*Ry*Rx per (layer,qubit), kron 4 of them into 16x16
// complex super-gates, and pack G^T (re), G^T (im), -G^T (im) directly in the
// WMMA f32 16x16x4 B-operand lane layout:
//   pack[kk*64 + lane*2 + v] = G[n][k],  n=lane&15, k=4*kk+2*(lane>>4)+v
// ws layout: (layer*3+g)*768 floats -> [Br(256) | Bi(256) | Bneg(256)]
// ---------------------------------------------------------------------------
__global__ void qgr_setup(const float* __restrict__ ax, const float* __restrict__ ay,
                          const float* __restrict__ az, float* __restrict__ bpack) {
    __shared__ float U[DEPTH * NQ][2][2][2];   // fused 2x2 complex gates
    const int t = threadIdx.x;
    if (t < DEPTH * NQ) {
        const float x = 0.5f * ax[t], y = 0.5f * ay[t], z = 0.5f * az[t];
        const float cx = cosf(x), sx = sinf(x);
        const float cy = cosf(y), sy = sinf(y);
        const float cz = cosf(z), sz = sinf(z);
        // M1 = Ry * Rx   (Rx = [[c,-i s],[-i s,c]], Ry = [[cy,-sy],[sy,cy]])
        const float m00r =  cy * cx, m00i =  sy * sx;
        const float m01r = -sy * cx, m01i = -cy * sx;
        const float m10r =  sy * cx, m10i = -cy * sx;
        const float m11r =  cy * cx, m11i = -sy * sx;
        // U = Rz * M1 ; Rz = diag(e^{-iz}, e^{+iz})
        U[t][0][0][0] = cz * m00r + sz * m00i;  U[t][0][0][1] = cz * m00i - sz * m00r;
        U[t][0][1][0] = cz * m01r + sz * m01i;  U[t][0][1][1] = cz * m01i - sz * m01r;
        U[t][1][0][0] = cz * m10r - sz * m10i;  U[t][1][0][1] = cz * m10i + sz * m10r;
        U[t][1][1][0] = cz * m11r - sz * m11i;  U[t][1][1][1] = cz * m11i + sz * m11r;
    }
    __syncthreads();
    const int mo = t >> 4;      // row n  of G (output sub-index)
    const int mi = t & 15;      // col k  of G (input  sub-index)
    for (int layer = 0; layer < DEPTH; ++layer) {
        for (int g = 0; g < 3; ++g) {
            float gr = 1.f, gi = 0.f;
#pragma unroll
            for (int jj = 0; jj < 4; ++jj) {        // qubit 4g+jj <-> sub-bit (3-jj)
                const int q  = 4 * g + jj;
                const int br = (mo >> (3 - jj)) & 1;
                const int bc = (mi >> (3 - jj)) & 1;
                const float ur = U[layer * NQ + q][br][bc][0];
                const float ui = U[layer * NQ + q][br][bc][1];
                const float nr = gr * ur - gi * ui;
                gi = gr * ui + gi * ur;
                gr = nr;
            }
            const int kk = mi >> 2, kh = (mi >> 1) & 1, v = mi & 1;
            const int lane = mo + 16 * kh;
            float* bse = bpack + (layer * 3 + g) * 768;
            bse[       kk * 64 + lane * 2 + v] =  gr;   // Br
            bse[256 +  kk * 64 + lane * 2 + v] =  gi;   // Bi
            bse[512 +  kk * 64 + lane * 2 + v] = -gi;   // -Bi
        }
    }
}

// ---------------------------------------------------------------------------
// Apply one 16x16 complex super-gate to the whole state in LDS via WMMA.
// P = stride of the 4-bit sub-index in the flat state index (1, 16, or 256).
// State element (r,k): addr = (r/P)*(16*P) + (r%P) + k*P,   r in [0,256).
// Per wave: 2 M-tiles of 16 rows. Y = X * G^T (complex) as 4 real GEMMs.
// ---------------------------------------------------------------------------
template <int P>
__device__ __forceinline__ void apply_group(float* sRe, float* sIm,
                                            const float* __restrict__ bp,
                                            int wave, int lane) {
    const int m  = lane & 15;     // A row within tile / D column
    const int kh = lane >> 4;     // K half-select
    v2f Br[4], Bi[4], Bn[4];
    const v2f* pb = (const v2f*)bp;
#pragma unroll
    for (int kk = 0; kk < 4; ++kk) {
        Br[kk] = pb[       kk * 32 + lane];
        Bi[kk] = pb[128 +  kk * 32 + lane];
        Bn[kk] = pb[256 +  kk * 32 + lane];
    }
#pragma unroll
    for (int tile = 0; tile < 2; ++tile) {
        const int r0   = (wave * 2 + tile) * 16;
        const int r    = r0 + m;
        const int base = (r / P) * (16 * P) + (r % P);
        v2f Are[4], Aim[4];
#pragma unroll
        for (int kk = 0; kk < 4; ++kk) {
            const int k0 = 4 * kk + 2 * kh;
            Are[kk].x = sRe[base +  k0      * P];
            Are[kk].y = sRe[base + (k0 + 1) * P];
            Aim[kk].x = sIm[base +  k0      * P];
            Aim[kk].y = sIm[base + (k0 + 1) * P];
        }
        v8f Dre = {0.f, 0.f, 0.f, 0.f, 0.f, 0.f, 0.f, 0.f};
        v8f Dim = {0.f, 0.f, 0.f, 0.f, 0.f, 0.f, 0.f, 0.f};
#pragma unroll
        for (int kk = 0; kk < 4; ++kk) {
            Dre = __builtin_amdgcn_wmma_f32_16x16x4_f32(false, Are[kk], false, Br[kk], (short)0, Dre, false, false);
            Dre = __builtin_amdgcn_wmma_f32_16x16x4_f32(false, Aim[kk], false, Bn[kk], (short)0, Dre, false, false);
            Dim = __builtin_amdgcn_wmma_f32_16x16x4_f32(false, Are[kk], false, Bi[kk], (short)0, Dim, false, false);
            Dim = __builtin_amdgcn_wmma_f32_16x16x4_f32(false, Aim[kk], false, Br[kk], (short)0, Dim, false, false);
        }
        // D layout: vgpr v, M = v + 8*kh, N = m
#pragma unroll
        for (int v = 0; v < 8; ++v) {
            const int rr = r0 + v + 8 * kh;
            const int b2 = (rr / P) * (16 * P) + (rr % P);
            sRe[b2 + m * P] = Dre[v];
            sIm[b2 + m * P] = Dim[v];
        }
    }
}

// ---------------------------------------------------------------------------
// Main kernel: one block per batch row. State lives in LDS (re/im f32).
// ---------------------------------------------------------------------------
__global__ void __launch_bounds__(THREADS)
qgr_main(const float* __restrict__ in, const float* __restrict__ bpack,
         float* __restrict__ out) {
    __shared__ float sRe[DIM];
    __shared__ float sIm[DIM];
    __shared__ float red1[THREADS];
    __shared__ float red2[THREADS];

    const int t    = threadIdx.x;
    const int b    = blockIdx.x;
    const int wave = t >> 5;
    const int lane = t & 31;

    // ---- amplitude encoding -------------------------------------------------
    const float* x = in + (size_t)b * DIM;
    float s1 = 0.f, s2 = 0.f;
#pragma unroll
    for (int j = 0; j < PERTH; ++j) {
        const int i = j * THREADS + t;
        const float w = fabsf(x[i]);
        sRe[i] = w;
        sIm[i] = 0.f;
        s1 += w;
        s2 += w * w;
    }
    red1[t] = s1; red2[t] = s2;
    __syncthreads();
    for (int off = THREADS / 2; off > 0; off >>= 1) {
        if (t < off) { red1[t] += red1[t + off]; red2[t] += red2[t + off]; }
        __syncthreads();
    }
    s1 = red1[0]; s2 = red2[0];
    const bool  uni  = !(s1 > 1e-8f);
    const float norm = uni ? 1.f : sqrtf(s2 / s1);          // sqrt(sum wn^2)
    const float invd = (norm > 1e-8f) ? (1.f / fmaxf(norm, 1e-8f)) : 1.f;
    const float fac  = uni ? 0.f : (1.f / sqrtf(s1)) * invd;
    const float cst  = uni ? (1.f / 64.f) * invd : 0.f;
#pragma unroll
    for (int j = 0; j < PERTH; ++j) {
        const int i = j * THREADS + t;
        sRe[i] = uni ? cst : sRe[i] * fac;
    }
    __syncthreads();

    // ---- circuit ------------------------------------------------------------
    for (int layer = 0; layer < DEPTH; ++layer) {
        const float* bl = bpack + layer * 3 * 768;
        apply_group<256>(sRe, sIm, bl,        wave, lane);  // qubits 0-3  (bits 11-8)
        __syncthreads();
        apply_group<16 >(sRe, sIm, bl + 768,  wave, lane);  // qubits 4-7  (bits 7-4)
        __syncthreads();
        apply_group<1  >(sRe, sIm, bl + 1536, wave, lane);  // qubits 8-11 (bits 3-0)
        __syncthreads();
        // fused CNOT ladder: scatter through registers
        float tr[PERTH], ti[PERTH];
        int   pj[PERTH];
#pragma unroll
        for (int j = 0; j < PERTH; ++j) {
            const int i = j * THREADS + t;
            tr[j] = sRe[i];
            ti[j] = sIm[i];
            pj[j] = (int)permL((unsigned)i);
        }
        __syncthreads();
#pragma unroll
        for (int j = 0; j < PERTH; ++j) {
            sRe[pj[j]] = tr[j];
            sIm[pj[j]] = ti[j];
        }
        __syncthreads();
    }

    // ---- measurement: sqrt(|amp|^2) = |amp| --------------------------------
    float* o = out + (size_t)b * DIM;
#pragma unroll
    for (int j = 0; j < PERTH; ++j) {
        const int i = j * THREADS + t;
        const float re = sRe[i], im = sIm[i];
        o[i] = sqrtf(re * re + im * im);
    }
}

// ---------------------------------------------------------------------------
extern "C" void kernel_launch(void* const* d_in, const int* in_sizes, int n_in,
                              void* d_out, int out_size, void* d_ws, size_t ws_size,
                              hipStream_t stream) {
    (void)in_sizes; (void)n_in; (void)out_size; (void)ws_size;
    const float* input = (const float*)d_in[0];
    const float* ax    = (const float*)d_in[1];
    const float* ay    = (const float*)d_in[2];
    const float* az    = (const float*)d_in[3];
    float* bpack = (float*)d_ws;                // 12 * 768 * 4 = 36,864 bytes used

    qgr_setup<<<1, THREADS, 0, stream>>>(ax, ay, az, bpack);
    qgr_main<<<NBATCH, THREADS, 0, stream>>>(input, bpack, (float*)d_out);
}